// JointEdgeSegLoss_85607288144418
// MI455X (gfx1250) — compile-verified
//
#include <hip/hip_runtime.h>
#include <hip/hip_bf16.h>

typedef __attribute__((ext_vector_type(2))) float v2f;
typedef __attribute__((ext_vector_type(4))) float v4f;
typedef __attribute__((ext_vector_type(4))) int   v4i;
typedef __attribute__((ext_vector_type(8))) float v8f;

#define C_CLS 19
#define HW (768 * 768)            // 589824
#define NIMG 4
#define BLK_PER_IMG 576           // 576*256*4 == 589824 exactly (1 iter, 4 px/thread)
#define THREADS 256
#define HIST_BLK_PER_IMG 288
#define HIST_ITERS 2              // 288*256*4*2 == 589824 exactly
#define EDGE_T 0.8f
#define L2E 1.4426950408889634f
#define LN2 0.6931471805599453f

// ---- ws layout (32-bit words) ----
// [0..75]     uint hist_seg[4][19]
// [76..151]   uint hist_att[4][19]
// [152]       uint pos_cnt
// [153]       uint neg_cnt
// [156..307]  float w2[4][19][2]  (interleaved {w_seg, w_att}; byte 624, 8B aligned)
// [308]       float bce_pos_w
// [309]       float bce_neg_w
// [312 ...]   float partials[2304][5]
#define WS_W2    156
#define WS_BCEW  308
#define WS_PART  312
#define NBLOCKS_MAIN (BLK_PER_IMG * NIMG)   // 2304

__global__ void k_zero(unsigned int* wsu) {
    int t = threadIdx.x;
    if (t < 154) wsu[t] = 0u;
}

__global__ void k_hist(const int* __restrict__ segmask,
                       const float* __restrict__ edgein,
                       const int* __restrict__ edgemask,
                       unsigned int* __restrict__ wsu) {
    __shared__ unsigned int lh[2 * C_CLS + 2];
    int tid = threadIdx.x;
    int img = blockIdx.y;
    if (tid < 2 * C_CLS + 2) lh[tid] = 0u;
    __syncthreads();

    size_t base = (size_t)img * HW;
    for (int it = 0; it < HIST_ITERS; ++it) {
        int p4 = (it * (HIST_BLK_PER_IMG * THREADS) + blockIdx.x * THREADS + tid) * 4;
        v4i cls4 = *(const v4i*)(segmask + base + p4);
        v4f e4   = *(const v4f*)(edgein + base + p4);
        v4i tm4  = *(const v4i*)(edgemask + base + p4);
        #pragma unroll
        for (int j = 0; j < 4; ++j) {
            int cls = min(max(cls4[j], 0), C_CLS - 1);
            atomicAdd(&lh[cls], 1u);
            if (e4[j] > EDGE_T) atomicAdd(&lh[C_CLS + cls], 1u);
            if (tm4[j] == 1)      atomicAdd(&lh[2 * C_CLS + 0], 1u);
            else if (tm4[j] == 0) atomicAdd(&lh[2 * C_CLS + 1], 1u);
        }
    }
    __syncthreads();
    if (tid < C_CLS) {
        atomicAdd(&wsu[img * C_CLS + tid], lh[tid]);
        atomicAdd(&wsu[76 + img * C_CLS + tid], lh[C_CLS + tid]);
    } else if (tid == C_CLS) {
        atomicAdd(&wsu[152], lh[2 * C_CLS + 0]);
        atomicAdd(&wsu[153], lh[2 * C_CLS + 1]);
    }
}

__global__ void k_weights(unsigned int* wsu) {
    float* wsf = (float*)wsu;
    int t = threadIdx.x;
    if (t < 152) {
        int which = t / 76;          // 0 = seg, 1 = att
        int idx = t % 76;            // n*19 + c
        int n = idx / 19;
        const unsigned int* h = wsu + which * 76 + n * C_CLS;
        float sum = 0.0f;
        for (int c = 0; c < C_CLS; ++c) sum += (float)h[c];
        unsigned int b = wsu[which * 76 + idx];
        // w = (bins!=0)*(1 - bins/sum) + 1
        float w = (b != 0u) ? (2.0f - (float)b / sum) : 1.0f;
        wsf[WS_W2 + idx * 2 + which] = w;
    } else if (t == 152) {
        float pos = (float)wsu[152], neg = (float)wsu[153];
        float s = pos + neg;
        wsf[WS_BCEW + 0] = neg / s;  // weight at positive pixels
        wsf[WS_BCEW + 1] = pos / s;  // weight at negative pixels
    }
}

__device__ __forceinline__ float hsum8(v8f d) {
    return ((d[0] + d[1]) + (d[2] + d[3])) + ((d[4] + d[5]) + (d[6] + d[7]));
}

__global__ void __launch_bounds__(THREADS)
k_main(const float* __restrict__ segin,
       const float* __restrict__ edgein,
       const int* __restrict__ segmask,
       const int* __restrict__ edgemask,
       float* __restrict__ wsf) {
    __shared__ v2f sw[C_CLS];       // {w_seg, w_att} for this image
    __shared__ float sbw[2];        // bce weights
    __shared__ float red[8][5][2];  // [wave][acc][half]

    const int tid = threadIdx.x;
    const int img = blockIdx.y;
    if (tid < C_CLS) sw[tid] = *(const v2f*)(wsf + WS_W2 + (img * C_CLS + tid) * 2);
    if (tid == C_CLS) { sbw[0] = wsf[WS_BCEW]; sbw[1] = wsf[WS_BCEW + 1]; }
    __syncthreads();

    const size_t pbase = (size_t)img * HW;
    const float* seg = segin + (size_t)img * C_CLS * HW;
    const int p4 = (blockIdx.x * THREADS + tid) * 4;   // 4 consecutive pixels / thread

    // ---- b128 streaming loads: 19 channel quads + masks/edge ----
    v4f v[C_CLS];
    #pragma unroll
    for (int c = 0; c < C_CLS; ++c)
        v[c] = *(const v4f*)(seg + (size_t)c * HW + p4);

    v4f m = v[0];
    #pragma unroll
    for (int c = 1; c < C_CLS; ++c) {
        #pragma unroll
        for (int j = 0; j < 4; ++j) m[j] = fmaxf(m[j], v[c][j]);
    }

    v4i cls4 = *(const v4i*)(segmask + pbase + p4);
    #pragma unroll
    for (int j = 0; j < 4; ++j) cls4[j] = min(max(cls4[j], 0), C_CLS - 1);

    // exp-sum + branchless target-logit select (no dynamic register indexing)
    v4f s = {0.f, 0.f, 0.f, 0.f};
    v4f xsel = v[0];
    #pragma unroll
    for (int c = 0; c < C_CLS; ++c) {
        #pragma unroll
        for (int j = 0; j < 4; ++j) {
            s[j] += exp2f((v[c][j] - m[j]) * L2E);
            xsel[j] = (cls4[j] == c) ? v[c][j] : xsel[j];
        }
    }
    v4f lse, lp;
    #pragma unroll
    for (int j = 0; j < 4; ++j) {
        lse[j] = m[j] + log2f(s[j]) * LN2;
        lp[j] = xsel[j] - lse[j];
    }

    v4f e4  = *(const v4f*)(edgein + pbase + p4);
    v4i tm4 = *(const v4i*)(edgemask + pbase + p4);

    float a_ns = 0.0f, a_ds = 0.0f;   // seg num/den
    float a_na = 0.0f, a_da = 0.0f;   // att num/den
    float a_wb = 0.0f;                // weighted bce sum
    #pragma unroll
    for (int j = 0; j < 4; ++j) {
        v2f w2 = sw[cls4[j]];
        a_ns -= w2[0] * lp[j];
        a_ds += w2[0];
        float wa = (e4[j] > EDGE_T) ? w2[1] : 0.0f;
        a_na -= wa * lp[j];
        a_da += wa;
        float tf = (float)tm4[j];
        float bce = fmaxf(e4[j], 0.0f) - e4[j] * tf + log1pf(expf(-fabsf(e4[j])));
        float wp = (tm4[j] == 1) ? sbw[0] : ((tm4[j] == 0) ? sbw[1] : 0.0f);
        a_wb += wp * bce;
    }

    // ---- wave-level reduction via V_WMMA_F32_16X16X4_F32 ----
    // A(16x4): lanes 0-15 give A[m][0],A[m][1]; lanes 16-31 give A[m][2],A[m][3].
    // B(4x16): VGPR0 = rows {0,2} (lanes 0-15 / 16-31), VGPR1 = rows {1,3}.
    //   D[m][0] = accX_m + accX_{m+16},  D[m][1] = accY_m + accY_{m+16}
    const int lid = tid & 31;
    const int wid = tid >> 5;
    v2f bsel;
    bsel[0] = (lid == 0 || lid == 16) ? 1.0f : 0.0f;  // rows 0,2 -> col 0
    bsel[1] = (lid == 1 || lid == 17) ? 1.0f : 0.0f;  // rows 1,3 -> col 1
    v8f cz = {0.f, 0.f, 0.f, 0.f, 0.f, 0.f, 0.f, 0.f};

    v2f A0; A0[0] = a_ns; A0[1] = a_ds;
    v2f A1; A1[0] = a_na; A1[1] = a_da;
    v2f A2; A2[0] = a_wb; A2[1] = 0.0f;
    v8f d0 = __builtin_amdgcn_wmma_f32_16x16x4_f32(false, A0, false, bsel, (short)0, cz, false, false);
    v8f d1 = __builtin_amdgcn_wmma_f32_16x16x4_f32(false, A1, false, bsel, (short)0, cz, false, false);
    v8f d2 = __builtin_amdgcn_wmma_f32_16x16x4_f32(false, A2, false, bsel, (short)0, cz, false, false);
    float r0 = hsum8(d0), r1 = hsum8(d1), r2 = hsum8(d2);

    // Col 0 lives in lanes 0 and 16 (M=0..7 / 8..15); col 1 in lanes 1 and 17.
    int half = (lid >= 16) ? 1 : 0;
    if (lid == 0 || lid == 16) {
        red[wid][0][half] = r0;   // num_seg partial
        red[wid][2][half] = r1;   // num_att partial
        red[wid][4][half] = r2;   // wbce partial
    } else if (lid == 1 || lid == 17) {
        red[wid][1][half] = r0;   // den_seg partial
        red[wid][3][half] = r1;   // den_att partial
    }
    __syncthreads();

    if (tid < 5) {
        float sacc = 0.0f;
        for (int w = 0; w < 8; ++w)
            for (int h = 0; h < 2; ++h) sacc += red[w][tid][h];
        int lin = img * BLK_PER_IMG + blockIdx.x;
        wsf[WS_PART + lin * 5 + tid] = sacc;
    }
}

__global__ void k_final(const float* __restrict__ wsf, float* __restrict__ out) {
    // 17 slots x 32 lanes; slots 0..15 = (img, acc 0..3); slot 16 = wbce total.
    __shared__ float lred[544];
    __shared__ float slot[17];
    int tid = threadIdx.x;
    int s = tid >> 5, lane = tid & 31;
    const float* part = wsf + WS_PART;
    float acc = 0.0f;
    if (s < 16) {
        int img = s >> 2, a = s & 3;
        for (int b = lane; b < BLK_PER_IMG; b += 32)
            acc += part[(img * BLK_PER_IMG + b) * 5 + a];
    } else {
        for (int b = lane; b < NBLOCKS_MAIN; b += 32)
            acc += part[b * 5 + 4];
    }
    lred[tid] = acc;
    __syncthreads();
    if (tid < 17) {
        float sum = 0.0f;
        for (int l = 0; l < 32; ++l) sum += lred[tid * 32 + l];
        slot[tid] = sum;
    }
    __syncthreads();
    if (tid == 0) {
        float loss = 0.0f;
        for (int n = 0; n < NIMG; ++n) {
            float ns = slot[n * 4 + 0], ds = slot[n * 4 + 1];
            float na = slot[n * 4 + 2], da = slot[n * 4 + 3];
            loss += 1.0f * (ns / ds);
            loss += 0.1f * (na / da);
        }
        loss += 0.3f * (slot[16] / (float)((size_t)NIMG * HW));
        out[0] = loss;
    }
}

extern "C" void kernel_launch(void* const* d_in, const int* in_sizes, int n_in,
                              void* d_out, int out_size, void* d_ws, size_t ws_size,
                              hipStream_t stream) {
    const float* segin   = (const float*)d_in[0];
    const float* edgein  = (const float*)d_in[1];
    const int*   segmask = (const int*)d_in[2];
    const int*   edgemask= (const int*)d_in[3];
    unsigned int* wsu = (unsigned int*)d_ws;
    float* wsf = (float*)d_ws;
    float* out = (float*)d_out;

    k_zero<<<1, 256, 0, stream>>>(wsu);
    k_hist<<<dim3(HIST_BLK_PER_IMG, NIMG), THREADS, 0, stream>>>(segmask, edgein, edgemask, wsu);
    k_weights<<<1, 256, 0, stream>>>(wsu);
    k_main<<<dim3(BLK_PER_IMG, NIMG), THREADS, 0, stream>>>(segin, edgein, segmask, edgemask, wsf);
    k_final<<<1, 544, 0, stream>>>(wsf, out);
}